// BIMM1D_11441792876623
// MI455X (gfx1250) — compile-verified
//
#include <hip/hip_runtime.h>
#include <hip/hip_bf16.h>

// CDNA5 / gfx1250 kernel for the Gaussian-mixture NLL reference.
//
// The (u - In)^2 quadratic AND the exp-argument scaling are both computed on
// the matrix pipe via V_WMMA_F32_16X16X4_F32:
//   A row = s * [In^2, -2*In, 1, 0],  B col = [1, u, u^2, 0],
//   s = -0.5/sn^2 * log2(e)
// so D = s*(u-In)^2 is directly the exp2 argument. The VALU then does only
// one v_exp_f32 + one add per matrix element — the minimum possible, since
// this kernel is transcendental-bound (201M Gaussian evals, ~1 MB of input).

typedef float v2f __attribute__((ext_vector_type(2)));
typedef float v8f __attribute__((ext_vector_type(8)));

#define SQRT2_F     1.41421356237309504880f
#define TWO_PI_F    6.28318530717958647692f
#define LOG2E_F     1.44269504088896340736f
#define LN2_F       0.69314718055994530942f

#if defined(__HIP_DEVICE_COMPILE__) && !__has_builtin(__builtin_amdgcn_wmma_f32_16x16x4_f32)
#error "wmma_f32_16x16x4_f32 builtin not available on this toolchain (device pass)"
#endif

// ---------------------------------------------------------------------------
// Kernel A: precompute per-component A-row coefficients, pre-scaled by
//   s = (-0.5/sn^2) * log2(e):   table[c] = { s*In^2, s*(-2*In) }
// plus scaled mixture weights; zero the output accumulator.
// ---------------------------------------------------------------------------
__global__ void bimm_precompute_kernel(const float* __restrict__ eps,      // (6, N)
                                       const float* __restrict__ Iphase,   // (4,)
                                       const float* __restrict__ sigma_b,  // (1,)
                                       const float* __restrict__ sigma_n,  // (1,)
                                       const float* __restrict__ dpar,     // (1,)
                                       const float* __restrict__ W,        // (10,)
                                       float2* __restrict__ table,         // (6*N,)
                                       float* __restrict__ wIf,            // (6,)
                                       float* __restrict__ wInt,           // (4,)
                                       float* __restrict__ out,            // (1,) zeroed
                                       int N)
{
    const int tid = threadIdx.x;
    const float sb = sigma_b[0];
    const float dd = dpar[0];
    const float sn = sigma_n[0];
    const float s  = (-0.5f / (sn * sn)) * LOG2E_F;   // exp2-domain quadratic scale
    // triu(k=1) pair order for 4 phases
    const int iaTab[6] = {0, 0, 0, 1, 1, 2};
    const int ibTab[6] = {1, 2, 3, 2, 3, 3};

    const int total = 6 * N;
    for (int c = tid; c < total; c += blockDim.x) {
        const int p = c / N;
        const int n = c - p * N;
        const float Ia = Iphase[iaTab[p]];
        const float Ib = Iphase[ibTab[p]];
        const float x  = eps[p * N + n] * (2.0f * dd * sb) - dd * sb;
        const float In = 0.5f * (erff(x / (SQRT2_F * sb)) + 1.0f) * (Ib - Ia) + Ia;
        table[c] = make_float2(s * In * In, s * (-2.0f * In));
    }

    if (tid == 0) {
        // softmax over the 10 mixture logits
        float m = W[0];
        for (int k = 1; k < 10; ++k) m = fmaxf(m, W[k]);
        float e[10];
        float den = 0.0f;
        for (int k = 0; k < 10; ++k) { e[k] = __expf(W[k] - m); den += e[k]; }
        // exp(c) with c = -log(sn) - 0.5*log(2*pi)  ==  1 / (sn * sqrt(2*pi))
        const float expc = 1.0f / (sn * sqrtf(TWO_PI_F));
        const float inv  = expc / den;
        for (int k = 0; k < 4; ++k) wInt[k] = e[k] * inv;                  // interior
        for (int p = 0; p < 6; ++p) wIf[p]  = e[4 + p] * inv / (float)N;  // interface (MC avg)
        out[0] = 0.0f;
    }
}

// ---------------------------------------------------------------------------
// Kernel B: one wave32 per 16-point u-tile.
//   v_wmma_f32_16x16x4_f32 produces D = s*(u_j - In_i)^2 directly;
//   per element: one v_exp_f32 (exp2) + one add. Then log + wave reduction.
// ---------------------------------------------------------------------------
__global__ void __launch_bounds__(256)
bimm_nll_kernel(const float* __restrict__ u,        // (M,)
                const float* __restrict__ Iphase,   // (4,)
                const float* __restrict__ sigma_n,  // (1,)
                const float2* __restrict__ table,   // (6*N,)
                const float* __restrict__ wIf,      // (6,)
                const float* __restrict__ wInt,     // (4,)
                float* __restrict__ out,            // (1,)
                int M, int N)
{
    const int wave = threadIdx.x >> 5;
    const int lane = threadIdx.x & 31;
    const int tile = blockIdx.x * (blockDim.x >> 5) + wave;
    if (tile * 16 >= M) return;          // wave-uniform: EXEC stays all-ones for WMMA

    const int col = lane & 15;           // data-point column within the tile
    const int hi  = lane >> 4;           // hi half-wave carries K=2/3 constants
    const int gidx = tile * 16 + col;
    const bool inRange = (gidx < M);
    const float uj = u[inRange ? gidx : (M - 1)];

    const float sn = sigma_n[0];
    const float s  = (-0.5f / (sn * sn)) * LOG2E_F;  // same scale as the table

    // B fragment (4x16): VGPR0 = K0 (ones) / K2 (u^2), VGPR1 = K1 (u) / K3 (zeros)
    v2f bfrag;
    bfrag.x = hi ? uj * uj : 1.0f;
    bfrag.y = hi ? 0.0f    : uj;

    const v8f czero = {};

    // interior (pure Gaussian) components — 4 cheap VALU evals
    float total = 0.0f;
#pragma unroll
    for (int k = 0; k < 4; ++k) {
        const float t = uj - Iphase[k];
        total += wInt[k] * __builtin_amdgcn_exp2f(s * t * t);
    }

    // interface components: 6 pairs x (N/16) WMMA sub-tiles
    const int subT = N >> 4;
    for (int p = 0; p < 6; ++p) {
        float sum = 0.0f;
        for (int st = 0; st < subT; ++st) {
            const int cbase = p * N + st * 16;
            const float2 t2 = table[cbase + col];   // scaled coeffs for component row
            v2f afrag;                               // A (16x4): lo lanes K0/K1, hi lanes K2/K3
            afrag.x = hi ? s    : t2.x;              // K0 = s*In^2   | K2 = s
            afrag.y = hi ? 0.0f : t2.y;              // K1 = s*(-2In) | K3 = 0
            const v8f dmat = __builtin_amdgcn_wmma_f32_16x16x4_f32(
                /*neg_a=*/false, afrag, /*neg_b=*/false, bfrag,
                /*c_mod=*/(short)0, czero, /*reuse_a=*/false, /*reuse_b=*/false);
#pragma unroll
            for (int v = 0; v < 8; ++v)
                sum += __builtin_amdgcn_exp2f(dmat[v]);  // raw v_exp_f32, no extra muls
        }
        sum += __shfl_xor(sum, 16, 32);              // combine row halves: lane j <-> j+16
        total += wIf[p] * sum;
    }

    // lanes j and j+16 hold identical totals -> weight by 0.5 in the reduction
    // ln(total) = v_log_f32(total) * ln(2)
    float val = inRange ? 0.5f * LN2_F * __builtin_amdgcn_logf(total) : 0.0f;
#pragma unroll
    for (int off = 16; off >= 1; off >>= 1)
        val += __shfl_xor(val, off, 32);
    if (lane == 0)
        atomicAdd(out, -val / (float)M);
}

extern "C" void kernel_launch(void* const* d_in, const int* in_sizes, int n_in,
                              void* d_out, int out_size, void* d_ws, size_t ws_size,
                              hipStream_t stream) {
    const float* u   = (const float*)d_in[0];
    const float* eps = (const float*)d_in[1];
    const float* I   = (const float*)d_in[2];
    const float* sb  = (const float*)d_in[3];
    const float* sn  = (const float*)d_in[4];
    const float* dp  = (const float*)d_in[5];
    const float* W   = (const float*)d_in[6];
    // d_in[7] = n_MC_components (int scalar); N is also derivable from eps size
    const int M = in_sizes[0];
    const int N = in_sizes[1] / 6;

    float* out = (float*)d_out;
    char*  ws  = (char*)d_ws;
    float2* table = (float2*)ws;                               // 6*N float2
    float*  wIf   = (float*)(ws + sizeof(float2) * (size_t)(6 * N));
    float*  wInt  = wIf + 6;

    bimm_precompute_kernel<<<1, 256, 0, stream>>>(eps, I, sb, sn, dp, W,
                                                  table, wIf, wInt, out, N);

    const int wavesPerBlock = 8;                               // 256 threads = 8 wave32
    const int tiles  = (M + 15) / 16;
    const int blocks = (tiles + wavesPerBlock - 1) / wavesPerBlock;
    bimm_nll_kernel<<<blocks, 256, 0, stream>>>(u, I, sn, table, wIf, wInt,
                                                out, M, N);
}